// CrystalAttention_62216896250132
// MI455X (gfx1250) — compile-verified
//
#include <hip/hip_runtime.h>
#include <cstdint>

// CrystalAttention fused kernel for gfx1250 (MI455X).
// y = ( softmax_n( scales_n / (||x_m - p_n|| + 0.1) ) @ V ) @ W_out^T + b_out
// All three GEMMs run on V_WMMA_F32_16X16X4_F32 (fp32 matrix pipe).
// Parameters are pre-transposed once per call so every B-fragment load is
// lane-contiguous (L2-friendly); x tile is staged to LDS via async-load-to-LDS.

typedef __attribute__((ext_vector_type(2))) float v2f;
typedef __attribute__((ext_vector_type(4))) float f4;
typedef __attribute__((ext_vector_type(8))) float v8f;
typedef __attribute__((ext_vector_type(4))) int   v4i;
typedef __attribute__((address_space(1))) v4i     v4i_as1;
typedef __attribute__((address_space(3))) v4i     v4i_as3;

#define DIM      512              // embed dim
#define NNEU     1024             // neurons
#define MT       32               // row tile per block
#define XS       516              // padded LDS row stride for x/mid (516 % 64 == 4)
#define SS       1028             // padded LDS row stride for scores (1028 % 64 == 4)
#define NTHREADS 512              // 16 waves

// One-shot derived parameter buffers (recomputed deterministically every call).
__device__ float g_posT[DIM * NNEU];   // [k][n] = positions[n][k]   (2 MB)
__device__ float g_WoutT[DIM * DIM];   // [k][d] = W_out[d][k]       (1 MB)
__device__ float g_psq[NNEU];          // ||positions[n]||^2

#if defined(__AMDGCN__) && __has_builtin(__builtin_amdgcn_global_load_async_to_lds_b128) && __has_builtin(__builtin_amdgcn_s_wait_asynccnt)
#define HAVE_ASYNC_LDS 1
#else
#define HAVE_ASYNC_LDS 0
#endif

static __device__ __forceinline__ v8f wmma_f32(v2f a, v2f b, v8f c) {
    return __builtin_amdgcn_wmma_f32_16x16x4_f32(false, a, false, b, (short)0, c, false, false);
}

// ---------------- one-shot parameter prep kernels ----------------

__global__ __launch_bounds__(256) void transpose_pos_kernel(const float* __restrict__ pos) {
    __shared__ float tile[32][33];
    const int tc = blockIdx.x * 32;            // k origin (cols of pos)
    const int tr = blockIdx.y * 32;            // n origin (rows of pos)
    const int j  = threadIdx.x & 31;
    const int i0 = threadIdx.x >> 5;
    for (int i = i0; i < 32; i += 8)
        tile[i][j] = pos[(size_t)(tr + i) * DIM + tc + j];
    __syncthreads();
    for (int i = i0; i < 32; i += 8)
        g_posT[(size_t)(tc + i) * NNEU + tr + j] = tile[j][i];
}

__global__ __launch_bounds__(256) void transpose_wout_kernel(const float* __restrict__ W) {
    __shared__ float tile[32][33];
    const int tc = blockIdx.x * 32;
    const int tr = blockIdx.y * 32;
    const int j  = threadIdx.x & 31;
    const int i0 = threadIdx.x >> 5;
    for (int i = i0; i < 32; i += 8)
        tile[i][j] = W[(size_t)(tr + i) * DIM + tc + j];
    __syncthreads();
    for (int i = i0; i < 32; i += 8)
        g_WoutT[(size_t)(tc + i) * DIM + tr + j] = tile[j][i];
}

__global__ __launch_bounds__(256) void psq_kernel(const float* __restrict__ pos) {
    const int n = blockIdx.x * 256 + threadIdx.x;
    const f4* p4 = (const f4*)(pos + (size_t)n * DIM);
    float s = 0.f;
#pragma unroll 8
    for (int k = 0; k < DIM / 4; ++k) {
        f4 v = p4[k];
        s += v.x * v.x + v.y * v.y + v.z * v.z + v.w * v.w;
    }
    g_psq[n] = s;
}

// ---------------- main fused kernel ----------------

__global__ __launch_bounds__(NTHREADS) void crystal_fused_kernel(
    const float* __restrict__ x,       // [16384, 512]
    const float* __restrict__ scales,  // [1024]
    const float* __restrict__ values,  // [1024, 512]
    const float* __restrict__ bout,    // [512]
    float* __restrict__ out)           // [16384, 512]
{
    __shared__ float lds_s[MT * SS];       // scores -> exp(scores)   (~131 KB)
    __shared__ float lds_xm[MT * XS];      // x tile, later mid tile  (~66 KB)
    __shared__ float lds_xsq[MT];
    __shared__ float lds_red[MT][16];
    __shared__ float lds_rowmax[MT];
    __shared__ float lds_rowsum[MT];       // holds 1/rowsum

    const int tid   = threadIdx.x;
    const int wave  = tid >> 5;            // 0..15
    const int lane  = tid & 31;
    const int col   = lane & 15;
    const int khalf = lane >> 4;           // 0: K=0,1  1: K=2,3
    const int koff  = khalf * 2;
    const int m0    = blockIdx.x * MT;

    // ---- stage x tile [32 x 512] into LDS ----
#if HAVE_ASYNC_LDS
    for (int idx = tid; idx < MT * (DIM / 4); idx += NTHREADS) {
        const int r = idx >> 7;            // 128 float4 per row
        const int c = idx & 127;
        // global flat address == AS1 address; LDS flat address low 32 bits == LDS offset
        v4i_as1* gp = (v4i_as1*)(uintptr_t)(x + (size_t)(m0 + r) * DIM + c * 4);
        v4i_as3* lp = (v4i_as3*)(uint32_t)(uintptr_t)(lds_xm + r * XS + c * 4);
        __builtin_amdgcn_global_load_async_to_lds_b128(gp, lp, 0, 0);
    }
    __builtin_amdgcn_s_wait_asynccnt(0);
#else
    for (int idx = tid; idx < MT * (DIM / 4); idx += NTHREADS) {
        const int r = idx >> 7;
        const int c = idx & 127;
        *(f4*)(lds_xm + r * XS + c * 4) = *(const f4*)(x + (size_t)(m0 + r) * DIM + c * 4);
    }
#endif
    __syncthreads();

    // ---- ||x_m||^2 (two-level reduction; 512 threads cover 32 rows x 16 lanes) ----
    {
        const int r = tid >> 4, j = tid & 15;
        float s = 0.f;
        for (int c = j; c < DIM; c += 16) { float v = lds_xm[r * XS + c]; s += v * v; }
        lds_red[r][j] = s;
    }
    __syncthreads();
    if (tid < MT) {
        float s = 0.f;
#pragma unroll
        for (int j = 0; j < 16; ++j) s += lds_red[tid][j];
        lds_xsq[tid] = s;
    }
    __syncthreads();

    // ========== Phase 1: cross GEMM (32x1024, K=512) -> distance -> score ==========
    // wave owns 4 consecutive N-tiles; 2 M-subtiles => 8 live accumulators.
    {
        const int n0 = wave * 64;
        v8f c00 = {}, c01 = {}, c02 = {}, c03 = {};
        v8f c10 = {}, c11 = {}, c12 = {}, c13 = {};
#pragma unroll 2
        for (int k = 0; k < DIM; k += 4) {
            v2f a0, a1, b;
            a0.x = lds_xm[col * XS + k + koff];
            a0.y = lds_xm[col * XS + k + koff + 1];
            a1.x = lds_xm[(col + 16) * XS + k + koff];
            a1.y = lds_xm[(col + 16) * XS + k + koff + 1];
            const float* pr0 = g_posT + (size_t)(k + koff) * NNEU + n0 + col;  // lane-contiguous
            const float* pr1 = pr0 + NNEU;
            b.x = pr0[0];  b.y = pr1[0];
            c00 = wmma_f32(a0, b, c00);  c10 = wmma_f32(a1, b, c10);
            b.x = pr0[16]; b.y = pr1[16];
            c01 = wmma_f32(a0, b, c01);  c11 = wmma_f32(a1, b, c11);
            b.x = pr0[32]; b.y = pr1[32];
            c02 = wmma_f32(a0, b, c02);  c12 = wmma_f32(a1, b, c12);
            b.x = pr0[48]; b.y = pr1[48];
            c03 = wmma_f32(a0, b, c03);  c13 = wmma_f32(a1, b, c13);
        }
        auto emit = [&](const v8f& acc, int mbase, int n) {
            const float psq = g_psq[n];
            const float sc  = scales[n];
#pragma unroll
            for (int j = 0; j < 8; ++j) {
                const int m = mbase + j + khalf * 8;
                float d2 = fmaxf(lds_xsq[m] + psq - 2.0f * acc[j], 0.0f);
                lds_s[m * SS + n] = sc / (sqrtf(d2) + 0.1f);
            }
        };
        emit(c00,  0, n0 + col);      emit(c01,  0, n0 + 16 + col);
        emit(c02,  0, n0 + 32 + col); emit(c03,  0, n0 + 48 + col);
        emit(c10, 16, n0 + col);      emit(c11, 16, n0 + 16 + col);
        emit(c12, 16, n0 + 32 + col); emit(c13, 16, n0 + 48 + col);
    }
    __syncthreads();

    // ========== Softmax over N (normalization deferred into mid write) ==========
    {
        const int r = tid >> 4, j = tid & 15;
        float mx = -3.0e38f;
        for (int c = j; c < NNEU; c += 16) mx = fmaxf(mx, lds_s[r * SS + c]);
        lds_red[r][j] = mx;
    }
    __syncthreads();
    if (tid < MT) {
        float mx = -3.0e38f;
#pragma unroll
        for (int j = 0; j < 16; ++j) mx = fmaxf(mx, lds_red[tid][j]);
        lds_rowmax[tid] = mx;
    }
    __syncthreads();
    {
        const int r = tid >> 4, j = tid & 15;
        const float rm = lds_rowmax[r];
        float s = 0.f;
        for (int c = j; c < NNEU; c += 16) {
            const float e = __expf(lds_s[r * SS + c] - rm);
            lds_s[r * SS + c] = e;
            s += e;
        }
        lds_red[r][j] = s;
    }
    __syncthreads();
    if (tid < MT) {
        float s = 0.f;
#pragma unroll
        for (int j = 0; j < 16; ++j) s += lds_red[tid][j];
        lds_rowsum[tid] = 1.0f / s;
    }
    __syncthreads();

    // ========== Phase 2: mid = exp_scores @ values  (K = 1024) ==========
    // wave owns 32 D-columns; 2 M-subtiles x 2 N-tiles.
    const int d0 = wave * 32;
    {
        v8f m00 = {}, m01 = {}, m10 = {}, m11 = {};
#pragma unroll 2
        for (int k = 0; k < NNEU; k += 4) {
            v2f a0, a1, b;
            a0.x = lds_s[col * SS + k + koff];
            a0.y = lds_s[col * SS + k + koff + 1];
            a1.x = lds_s[(col + 16) * SS + k + koff];
            a1.y = lds_s[(col + 16) * SS + k + koff + 1];
            const float* v0 = values + (size_t)(k + koff) * DIM + d0 + col;   // lane-contiguous
            const float* v1 = v0 + DIM;
            b.x = v0[0];  b.y = v1[0];
            m00 = wmma_f32(a0, b, m00);  m10 = wmma_f32(a1, b, m10);
            b.x = v0[16]; b.y = v1[16];
            m01 = wmma_f32(a0, b, m01);  m11 = wmma_f32(a1, b, m11);
        }
        __syncthreads();                 // x tile dead; reuse lds_xm for mid
#pragma unroll
        for (int j = 0; j < 8; ++j) {
            const int ma = j + khalf * 8;
            const int mb = ma + 16;
            const float ia = lds_rowsum[ma];
            const float ib = lds_rowsum[mb];
            lds_xm[ma * XS + d0 + col     ] = m00[j] * ia;
            lds_xm[ma * XS + d0 + col + 16] = m01[j] * ia;
            lds_xm[mb * XS + d0 + col     ] = m10[j] * ib;
            lds_xm[mb * XS + d0 + col + 16] = m11[j] * ib;
        }
    }
    __syncthreads();

    // ========== Phase 3: y = mid @ Wout^T + b  (K = 512) ==========
    {
        v8f o00 = {}, o01 = {}, o10 = {}, o11 = {};
#pragma unroll 2
        for (int k = 0; k < DIM; k += 4) {
            v2f a0, a1, b;
            a0.x = lds_xm[col * XS + k + koff];
            a0.y = lds_xm[col * XS + k + koff + 1];
            a1.x = lds_xm[(col + 16) * XS + k + koff];
            a1.y = lds_xm[(col + 16) * XS + k + koff + 1];
            const float* w0 = g_WoutT + (size_t)(k + koff) * DIM + d0 + col;  // lane-contiguous
            const float* w1 = w0 + DIM;
            b.x = w0[0];  b.y = w1[0];
            o00 = wmma_f32(a0, b, o00);  o10 = wmma_f32(a1, b, o10);
            b.x = w0[16]; b.y = w1[16];
            o01 = wmma_f32(a0, b, o01);  o11 = wmma_f32(a1, b, o11);
        }
        const float b0 = bout[d0 + col];
        const float b1 = bout[d0 + col + 16];
#pragma unroll
        for (int j = 0; j < 8; ++j) {
            const int ma = j + khalf * 8;
            const int mb = ma + 16;
            out[(size_t)(m0 + ma) * DIM + d0 + col     ] = o00[j] + b0;
            out[(size_t)(m0 + ma) * DIM + d0 + col + 16] = o01[j] + b1;
            out[(size_t)(m0 + mb) * DIM + d0 + col     ] = o10[j] + b0;
            out[(size_t)(m0 + mb) * DIM + d0 + col + 16] = o11[j] + b1;
        }
    }
}

extern "C" void kernel_launch(void* const* d_in, const int* in_sizes, int n_in,
                              void* d_out, int out_size, void* d_ws, size_t ws_size,
                              hipStream_t stream) {
    const float* x      = (const float*)d_in[0];   // [8,2048,512]
    const float* pos    = (const float*)d_in[1];   // [1024,512]
    const float* scales = (const float*)d_in[2];   // [1024]
    const float* values = (const float*)d_in[3];   // [1024,512]
    const float* Wout   = (const float*)d_in[4];   // [512,512]
    const float* bout   = (const float*)d_in[5];   // [512]
    float* out = (float*)d_out;

    // One-shot derived-parameter prep (same stream => ordered before main kernel).
    transpose_pos_kernel <<<dim3(DIM / 32, NNEU / 32), 256, 0, stream>>>(pos);
    transpose_wout_kernel<<<dim3(DIM / 32, DIM  / 32), 256, 0, stream>>>(Wout);
    psq_kernel           <<<NNEU / 256, 256, 0, stream>>>(pos);

    const int M = in_sizes[0] / DIM;               // 16384 rows
    crystal_fused_kernel<<<M / MT, NTHREADS, 0, stream>>>(x, scales, values, bout, out);
}